// CategoryCrossAttention_81295140979264
// MI455X (gfx1250) — compile-verified
//
#include <hip/hip_runtime.h>
#include <hip/hip_bf16.h>

#define B   32
#define T   4096
#define CAT 128
#define NE  512
#define HS  64

#define ROWS_BLK 256                       // rows of x per flash block
#define NCHUNK   (T / ROWS_BLK)            // 16 partials per batch row

typedef float v2f __attribute__((ext_vector_type(2)));
typedef float v4f __attribute__((ext_vector_type(4)));
typedef float v8f __attribute__((ext_vector_type(8)));

// ---------------------------------------------------------------------------
// Kernel 1: q[b] = cat_emb[b] @ Wq ; r[b] = (Wk @ q[b]) * 1/sqrt(HS)
// (score[b,t] = x[b,t,:] . r[b,:] -- K projection folded away)
// ---------------------------------------------------------------------------
__global__ void k_qr(const float* __restrict__ cat_emb,
                     const float* __restrict__ Wq,
                     const float* __restrict__ Wk,
                     float* __restrict__ r) {
  const int b = blockIdx.x;
  const int tid = threadIdx.x;              // 256 threads
  __shared__ float q[HS];

  if (tid < HS) {
    float acc = 0.0f;
    for (int c = 0; c < CAT; ++c)
      acc += cat_emb[b * CAT + c] * Wq[c * HS + tid];
    q[tid] = acc;
  }
  __syncthreads();

  for (int n = tid; n < NE; n += 256) {
    float acc = 0.0f;
#pragma unroll
    for (int h = 0; h < HS; ++h)
      acc += Wk[n * HS + h] * q[h];
    r[b * NE + n] = acc * 0.125f;           // fold 1/sqrt(64) into r
  }
}

// ---------------------------------------------------------------------------
// Kernel 2: single-pass online-softmax accumulation (flash style).
// Each wave: for each of its rows, dot(x_row, r) -> wave-uniform score,
// then online update of (m, s, u~[n]) with the row still in registers.
// Per-block partial (m, s, u~[512]) written to workspace.
// x is read EXACTLY ONCE (non-temporal).
// ---------------------------------------------------------------------------
__global__ void k_flash(const float* __restrict__ x,
                        const float* __restrict__ r,
                        float* __restrict__ pm,
                        float* __restrict__ ps,
                        float* __restrict__ pu) {
  const int b = blockIdx.x;
  const int chunk = blockIdx.y;
  const int t0 = chunk * ROWS_BLK;
  const int wave = threadIdx.x >> 5;
  const int lane = threadIdx.x & 31;

  __shared__ float rs[NE];
  __shared__ float us[8][NE];               // per-wave u~ partials (16 KB)
  __shared__ float ms[8], ss[8];

  for (int i = threadIdx.x; i < NE; i += 256) rs[i] = r[b * NE + i];
  __syncthreads();

  // r slice held in registers (lane owns n = 4*(lane+32*i)..+3, i=0..3)
  v4f rreg[4];
#pragma unroll
  for (int i = 0; i < 4; ++i) rreg[i] = ((const v4f*)rs)[lane + 32 * i];

  v4f acc[4];
#pragma unroll
  for (int i = 0; i < 4; ++i) acc[i] = (v4f)0.0f;
  float m = -3.4e38f, s = 0.0f;

  const float* xb = x + (size_t)b * T * NE + (size_t)t0 * NE;
  const int rows_per_wave = ROWS_BLK / 8;   // 32

  for (int rr = 0; rr < rows_per_wave; ++rr) {
    const int t = wave * rows_per_wave + rr;
    const v4f* row = (const v4f*)(xb + (size_t)t * NE);

    v4f xv[4];
    float dot = 0.0f;
#pragma unroll
    for (int i = 0; i < 4; ++i) {
      xv[i] = __builtin_nontemporal_load(&row[lane + 32 * i]);
      dot += xv[i].x * rreg[i].x + xv[i].y * rreg[i].y +
             xv[i].z * rreg[i].z + xv[i].w * rreg[i].w;
    }
#pragma unroll
    for (int off = 16; off > 0; off >>= 1) dot += __shfl_xor(dot, off, 32);
    // dot is wave-uniform -> branch below is divergence-free

    float p;
    if (dot > m) {                          // rare after warm-up
      const float corr = __expf(m - dot);   // first iter: exp(-inf)=0
      s *= corr;
#pragma unroll
      for (int i = 0; i < 4; ++i) acc[i] *= corr;
      m = dot;
      p = 1.0f;
    } else {
      p = __expf(dot - m);
    }
    s += p;
#pragma unroll
    for (int i = 0; i < 4; ++i) acc[i] += p * xv[i];
  }

  // wave partial -> LDS
#pragma unroll
  for (int i = 0; i < 4; ++i) ((v4f*)us[wave])[lane + 32 * i] = acc[i];
  if (lane == 0) { ms[wave] = m; ss[wave] = s; }
  __syncthreads();

  // block merge (8 wave partials), 2 n-slots per thread
  float M = ms[0];
#pragma unroll
  for (int w = 1; w < 8; ++w) M = fmaxf(M, ms[w]);
  float f[8];
  float Sb = 0.0f;
#pragma unroll
  for (int w = 0; w < 8; ++w) { f[w] = __expf(ms[w] - M); Sb += f[w] * ss[w]; }

  const int n0 = threadIdx.x, n1 = threadIdx.x + 256;
  float u0 = 0.0f, u1 = 0.0f;
#pragma unroll
  for (int w = 0; w < 8; ++w) {
    u0 += f[w] * us[w][n0];
    u1 += f[w] * us[w][n1];
  }
  const int pidx = b * NCHUNK + chunk;
  if (threadIdx.x == 0) { pm[pidx] = M; ps[pidx] = Sb; }
  pu[(size_t)pidx * NE + n0] = u0;
  pu[(size_t)pidx * NE + n1] = u1;
}

// ---------------------------------------------------------------------------
// Kernel 3: merge the NCHUNK partials per batch row; emit normalized
// u[b,n] = (sum_t softmax_wei * x[b,t,n]).
// ---------------------------------------------------------------------------
__global__ void k_merge(const float* __restrict__ pm,
                        const float* __restrict__ ps,
                        const float* __restrict__ pu,
                        float* __restrict__ u) {
  const int b = blockIdx.x;
  const int tid = threadIdx.x;              // 256 threads, 2 n each

  float M = -3.4e38f;
#pragma unroll
  for (int c = 0; c < NCHUNK; ++c) M = fmaxf(M, pm[b * NCHUNK + c]);

  float f[NCHUNK];
  float S = 0.0f;
#pragma unroll
  for (int c = 0; c < NCHUNK; ++c) {
    f[c] = __expf(pm[b * NCHUNK + c] - M);
    S += f[c] * ps[b * NCHUNK + c];
  }
  const float invS = 1.0f / S;

  const int n0 = tid, n1 = tid + 256;
  float u0 = 0.0f, u1 = 0.0f;
#pragma unroll
  for (int c = 0; c < NCHUNK; ++c) {
    const size_t base = (size_t)(b * NCHUNK + c) * NE;
    u0 += f[c] * pu[base + n0];
    u1 += f[c] * pu[base + n1];
  }
  u[b * NE + n0] = u0 * invS;
  u[b * NE + n1] = u1 * invS;
}

// ---------------------------------------------------------------------------
// Kernel 4: WMMA head.  out[32x64] = u[32x512] @ Wv[512x64]
//           then y0[32x512] = out[32x64] @ Wp[64x512]
// V_WMMA_F32_16X16X4_F32, one 256-thread block (8 waves).
// A layout (16x4 f32): lanes 0-15 hold M=lane, K0/K1; lanes 16-31 hold K2/K3.
// B layout mirrors with N=lane&15.  C/D: VGPR j -> rows j (lanes<16), j+8.
// ---------------------------------------------------------------------------
__global__ void k_head(const float* __restrict__ u,
                       const float* __restrict__ Wv,
                       const float* __restrict__ Wp,
                       float* __restrict__ y0) {
  __shared__ float outs[32 * 64];           // 8 KB
  const int tid = threadIdx.x;
  const int wave = tid >> 5;
  const int lane = tid & 31;
  const int l16 = lane & 15;
  const int hi  = lane >> 4;                // 0: K0/K1, 1: K2/K3

  // ---- GEMM1: tile (mt, nt), one per wave --------------------------------
  {
    const int mt = wave >> 2;               // 0..1
    const int nt = wave & 3;                // 0..3
    const int m = mt * 16 + l16;
    const int n = nt * 16 + l16;
    v8f c = {0.f, 0.f, 0.f, 0.f, 0.f, 0.f, 0.f, 0.f};
    for (int k = 0; k < NE; k += 4) {
      const int ka = k + hi * 2;
      v2f a;  a.x = u[m * NE + ka];        a.y = u[m * NE + ka + 1];
      v2f bm; bm.x = Wv[ka * HS + n];      bm.y = Wv[(ka + 1) * HS + n];
      c = __builtin_amdgcn_wmma_f32_16x16x4_f32(false, a, false, bm,
                                                (short)0, c, false, false);
    }
#pragma unroll
    for (int j = 0; j < 8; ++j) {
      const int row = mt * 16 + j + hi * 8;
      outs[row * 64 + n] = c[j];
    }
  }
  __syncthreads();

  // ---- GEMM2: 2x32 tiles, 8 per wave -------------------------------------
  for (int i = 0; i < 8; ++i) {
    const int tI = wave + 8 * i;            // 0..63
    const int mt = tI >> 5;                 // 0..1
    const int nt = tI & 31;                 // 0..31
    const int m = mt * 16 + l16;
    const int n = nt * 16 + l16;
    v8f c = {0.f, 0.f, 0.f, 0.f, 0.f, 0.f, 0.f, 0.f};
    for (int k = 0; k < HS; k += 4) {
      const int ka = k + hi * 2;
      v2f a;  a.x = outs[m * 64 + ka];     a.y = outs[m * 64 + ka + 1];
      v2f bm; bm.x = Wp[ka * NE + n];      bm.y = Wp[(ka + 1) * NE + n];
      c = __builtin_amdgcn_wmma_f32_16x16x4_f32(false, a, false, bm,
                                                (short)0, c, false, false);
    }
#pragma unroll
    for (int j = 0; j < 8; ++j) {
      const int row = mt * 16 + j + hi * 8;
      y0[row * NE + n] = c[j];
    }
  }
}

// ---------------------------------------------------------------------------
// Kernel 5: LayerNorm over NE per batch row.
// ---------------------------------------------------------------------------
__global__ void k_ln(const float* __restrict__ y0,
                     const float* __restrict__ gamma,
                     const float* __restrict__ beta,
                     float* __restrict__ yln) {
  const int b = blockIdx.x;
  const int tid = threadIdx.x;              // 256 threads, 2 elems each
  __shared__ float red[256];

  const float a0 = y0[b * NE + tid];
  const float a1 = y0[b * NE + 256 + tid];

  red[tid] = a0 + a1;
  __syncthreads();
  for (int off = 128; off > 0; off >>= 1) {
    if (tid < off) red[tid] += red[tid + off];
    __syncthreads();
  }
  const float mu = red[0] * (1.0f / NE);
  __syncthreads();

  const float d0 = a0 - mu, d1 = a1 - mu;
  red[tid] = d0 * d0 + d1 * d1;
  __syncthreads();
  for (int off = 128; off > 0; off >>= 1) {
    if (tid < off) red[tid] += red[tid + off];
    __syncthreads();
  }
  const float inv = rsqrtf(red[0] * (1.0f / NE) + 1e-5f);

  yln[b * NE + tid]       = d0 * inv * gamma[tid]       + beta[tid];
  yln[b * NE + 256 + tid] = d1 * inv * gamma[256 + tid] + beta[256 + tid];
}

// ---------------------------------------------------------------------------
// Kernel 6: broadcast yln[b,:] to out[b, t, :] for all t (pure stream-out).
// Non-temporal 128-bit stores; 16 rows per block.
// ---------------------------------------------------------------------------
__global__ void k_bcast(const float* __restrict__ yln,
                        float* __restrict__ out) {
  const int b = blockIdx.x;
  const int t0 = blockIdx.y * 16;
  const int tid = threadIdx.x;
  const int f = tid & 127;                  // float4 index within row (NE/4)
  const int rsub = tid >> 7;                // 0..1

  const v4f val = ((const v4f*)(yln + b * NE))[f];
  v4f* ob = (v4f*)(out + (size_t)b * T * NE + (size_t)t0 * NE);
#pragma unroll
  for (int i = 0; i < 8; ++i) {
    __builtin_nontemporal_store(val, &ob[(size_t)(i * 2 + rsub) * 128 + f]);
  }
}

// ---------------------------------------------------------------------------
extern "C" void kernel_launch(void* const* d_in, const int* in_sizes, int n_in,
                              void* d_out, int out_size, void* d_ws, size_t ws_size,
                              hipStream_t stream) {
  const float* x       = (const float*)d_in[0];
  const float* cat_emb = (const float*)d_in[1];
  const float* Wq      = (const float*)d_in[2];
  const float* Wk      = (const float*)d_in[3];
  const float* Wv      = (const float*)d_in[4];
  const float* Wp      = (const float*)d_in[5];
  const float* gamma   = (const float*)d_in[6];
  const float* beta    = (const float*)d_in[7];
  float* out = (float*)d_out;

  float* ws  = (float*)d_ws;
  float* r   = ws;                          // [B, NE]            16384
  float* pm  = ws + 16384;                  // [B, NCHUNK]          512
  float* ps  = ws + 16896;                  // [B, NCHUNK]          512
  float* pu  = ws + 17408;                  // [B, NCHUNK, NE]   262144
  float* u   = ws + 279552;                 // [B, NE]            16384
  float* y0  = ws + 295936;                 // [B, NE]            16384
  float* yln = ws + 312320;                 // [B, NE]            16384

  k_qr    <<<dim3(B),           256, 0, stream>>>(cat_emb, Wq, Wk, r);
  k_flash <<<dim3(B, NCHUNK),   256, 0, stream>>>(x, r, pm, ps, pu);
  k_merge <<<dim3(B),           256, 0, stream>>>(pm, ps, pu, u);
  k_head  <<<dim3(1),           256, 0, stream>>>(u, Wv, Wp, y0);
  k_ln    <<<dim3(B),           256, 0, stream>>>(y0, gamma, beta, yln);
  k_bcast <<<dim3(B, T/16),     256, 0, stream>>>(yln, out);
}